// MultiheadFlashDiff2_68676527063267
// MI455X (gfx1250) — compile-verified
//
#include <hip/hip_runtime.h>
#include <hip/hip_bf16.h>
#include <math.h>

typedef __attribute__((ext_vector_type(2))) float v2f;
typedef __attribute__((ext_vector_type(8))) float v8f;

#define LAMBDA_INIT 0.7836057665316244f
#define EPSV 1e-5f

// ---- CDNA5 async global->LDS copy (ASYNCcnt) --------------------------------
__device__ __forceinline__ void async_load_b128(const float* gptr, void* lptr) {
  // Low 32 bits of a generic pointer into the shared aperture are the DS
  // byte address (ISA 10.2).
  unsigned int l = (unsigned int)(unsigned long long)lptr;
  asm volatile("global_load_async_to_lds_b128 %0, %1, off"
               :: "v"(l), "v"(gptr)
               : "memory");
}
__device__ __forceinline__ void wait_async0() {
  asm volatile("s_wait_asynccnt 0" ::: "memory");
}
__device__ __forceinline__ void wait_ds0() {
  asm volatile("s_wait_dscnt 0" ::: "memory");
}

// ---------------------------------------------------------------------------
// GEMM: C[M,N] = A[M,K] @ W[N,K]^T   (fp32 WMMA 16x16x4)
// 256 threads (8 waves), tile 128(M) x 64(N), K-chunk 16, double-buffered
// async global->LDS staging. Each wave owns a 16x64 strip.
// ---------------------------------------------------------------------------
__global__ __launch_bounds__(256) void proj_gemm(const float* __restrict__ A,
                                                 const float* __restrict__ W,
                                                 float* __restrict__ C,
                                                 int M, int N, int K) {
  __shared__ float As[2][128][20];  // pad 20 dwords: conflict-free frag reads
  __shared__ float Ws[2][64][20];
  const int t = threadIdx.x;
  const int wave = t >> 5, lane = t & 31;
  const int lane15 = lane & 15, kh = lane >> 4;  // kh: K-half of the wave
  const int rowBase = blockIdx.y * 128;
  const int colBase = blockIdx.x * 64;

  v8f acc[4];
#pragma unroll
  for (int nt = 0; nt < 4; ++nt)
#pragma unroll
    for (int i = 0; i < 8; ++i) acc[nt][i] = 0.0f;

  const int ar = t >> 1;        // 0..127
  const int ac = (t & 1) * 8;   // 0 or 8
  const int wr = t >> 2;        // 0..63
  const int wc = (t & 3) * 4;   // 0,4,8,12

  const float* Ap = A + (size_t)(rowBase + ar) * K + ac;
  const float* Wp = W + (size_t)(colBase + wr) * K + wc;

  auto issue_chunk = [&](int kc, int bufi) {
    async_load_b128(Ap + kc,     &As[bufi][ar][ac]);
    async_load_b128(Ap + kc + 4, &As[bufi][ar][ac + 4]);
    async_load_b128(Wp + kc,     &Ws[bufi][wr][wc]);
  };

  issue_chunk(0, 0);
  wait_async0();
  __syncthreads();

  int cur = 0;
  for (int kc = 0; kc < K; kc += 16) {
    if (kc + 16 < K) issue_chunk(kc + 16, cur ^ 1);  // overlap with compute
#pragma unroll
    for (int k0 = 0; k0 < 16; k0 += 4) {
      v2f a;  // A-layout: lanes 0-15 K=k0+{0,1}, lanes 16-31 K=k0+{2,3}
      a.x = As[cur][wave * 16 + lane15][k0 + 2 * kh];
      a.y = As[cur][wave * 16 + lane15][k0 + 2 * kh + 1];
#pragma unroll
      for (int nt = 0; nt < 4; ++nt) {
        v2f bfr;  // B[k][n] = W[colBase+n][k]
        bfr.x = Ws[cur][nt * 16 + lane15][k0 + 2 * kh];
        bfr.y = Ws[cur][nt * 16 + lane15][k0 + 2 * kh + 1];
        acc[nt] = __builtin_amdgcn_wmma_f32_16x16x4_f32(false, a, false, bfr,
                                                        (short)0, acc[nt],
                                                        false, false);
      }
    }
    wait_async0();
    __syncthreads();
    cur ^= 1;
  }

#pragma unroll
  for (int nt = 0; nt < 4; ++nt)
#pragma unroll
    for (int i = 0; i < 8; ++i) {
      int r = rowBase + wave * 16 + 8 * kh + i;  // C layout: VGPR i -> M=i / i+8
      int c = colBase + nt * 16 + lane15;
      C[(size_t)r * N + c] = acc[nt][i];
    }
}

// ---------------------------------------------------------------------------
// Interleaved RoPE over (B,S,2H,HD): one thread per even/odd pair.
// ---------------------------------------------------------------------------
__global__ __launch_bounds__(256) void rope_kernel(float* __restrict__ buf,
                                                   int S, int total) {
  int idx = blockIdx.x * blockDim.x + threadIdx.x;
  if (idx >= total) return;
  int p = idx & 31;            // pair index within HD/2 = 32
  int bt = idx >> 10;          // (b*S + t), since 2H*HD/2 = 32*32 = 1024
  int tpos = bt % S;
  float ang = powf(10000.0f, -(float)p * (1.0f / 31.0f));  // linspace(0,1,32)
  float f = (float)tpos * ang;
  float c = cosf(f), sn = sinf(f);
  float* base = buf + (size_t)idx * 2;
  float xe = base[0], xo = base[1];
  base[0] = xe * c - xo * sn;
  base[1] = xe * sn + xo * c;
}

// ---------------------------------------------------------------------------
// lambda = exp(sum lq1*lk1) - exp(sum lq2*lk2) + LAMBDA_INIT
// ---------------------------------------------------------------------------
__global__ void lam_kernel(const float* __restrict__ lq1, const float* __restrict__ lk1,
                           const float* __restrict__ lq2, const float* __restrict__ lk2,
                           float* __restrict__ out) {
  float s1 = 0.0f, s2 = 0.0f;
  for (int i = 0; i < 64; ++i) { s1 += lq1[i] * lk1[i]; s2 += lq2[i] * lk2[i]; }
  *out = expf(s1) - expf(s2) + LAMBDA_INIT;
}

// ---------------------------------------------------------------------------
// Flash attention (non-causal), fp32 WMMA, double-buffered async K/V tiles.
// grid = (S/128, H, B*2);   z = b*2 + component. One wave per 16 query rows.
// ---------------------------------------------------------------------------
__global__ __launch_bounds__(256) void attn_kernel(const float* __restrict__ qb,
                                                   const float* __restrict__ kb,
                                                   const float* __restrict__ vb,
                                                   float* __restrict__ o1,
                                                   float* __restrict__ o2) {
  const int S = 2048, E = 2048, H = 16, HD = 64;
  __shared__ float kt[2][16][68];    // K tile (16 keys x 64), pad 68
  __shared__ float vt[2][16][128];   // V tile (16 keys x 128)
  __shared__ float pt[8][16][20];    // per-wave P transpose patch, pad 20

  const int t = threadIdx.x;
  const int wave = t >> 5, lane = t & 31;
  const int lane15 = lane & 15, kh = lane >> 4;
  const int h = blockIdx.y;
  const int b = blockIdx.z >> 1;
  const int comp = blockIdx.z & 1;
  const int qrow0 = blockIdx.x * 128 + wave * 16;
  float* __restrict__ ob = comp ? o2 : o1;

  // Preload this wave's Q block (16x64) as 16 A-fragments.
  const float* qrow = qb + ((size_t)(b * S + qrow0 + lane15)) * E + (2 * h + comp) * HD;
  v2f qa[16];
#pragma unroll
  for (int ks = 0; ks < 16; ++ks) {
    qa[ks].x = qrow[4 * ks + 2 * kh];
    qa[ks].y = qrow[4 * ks + 2 * kh + 1];
  }

  float mrow[8], lrow[8];
  v8f o[8];
#pragma unroll
  for (int i = 0; i < 8; ++i) { mrow[i] = -INFINITY; lrow[i] = 0.0f; }
#pragma unroll
  for (int nt = 0; nt < 8; ++nt)
#pragma unroll
    for (int i = 0; i < 8; ++i) o[nt][i] = 0.0f;

  const float* kbase = kb + (size_t)b * S * E + (2 * h + comp) * HD;
  const float* vbase = vb + (size_t)b * S * E + h * 128;

  const int ldr = t >> 4;          // 0..15: tile row this thread stages
  const int kld = (t & 15) * 4;    // 4 floats of a K row
  const int vld = (t & 15) * 8;    // 8 floats of a V row

  auto issue_tiles = [&](int j, int bufi) {
    const float* kp = kbase + (size_t)(j * 16 + ldr) * E + kld;
    async_load_b128(kp, &kt[bufi][ldr][kld]);
    const float* vp = vbase + (size_t)(j * 16 + ldr) * E + vld;
    async_load_b128(vp,     &vt[bufi][ldr][vld]);
    async_load_b128(vp + 4, &vt[bufi][ldr][vld + 4]);
  };

  issue_tiles(0, 0);
  wait_async0();
  __syncthreads();

  int cur = 0;
  const int nJ = S / 16;
  for (int j = 0; j < nJ; ++j) {
    if (j + 1 < nJ) issue_tiles(j + 1, cur ^ 1);  // overlap with compute

    // S = Q @ K^T  (16x16, K=64 -> 16 WMMAs)
    v8f s;
#pragma unroll
    for (int i = 0; i < 8; ++i) s[i] = 0.0f;
#pragma unroll
    for (int ks = 0; ks < 16; ++ks) {
      v2f bfr;  // B[k][n] = K[n][k]
      bfr.x = kt[cur][lane15][4 * ks + 2 * kh];
      bfr.y = kt[cur][lane15][4 * ks + 2 * kh + 1];
      s = __builtin_amdgcn_wmma_f32_16x16x4_f32(false, qa[ks], false, bfr,
                                                (short)0, s, false, false);
    }

    // Online softmax. Lanes 0-15 own rows 0..7, lanes 16-31 rows 8..15.
    float pv[8], alpha[8];
#pragma unroll
    for (int i = 0; i < 8; ++i) {
      float sv = s[i] * 0.125f;  // HD^-0.5
      float rm = sv;
      rm = fmaxf(rm, __shfl_xor(rm, 1, 32));
      rm = fmaxf(rm, __shfl_xor(rm, 2, 32));
      rm = fmaxf(rm, __shfl_xor(rm, 4, 32));
      rm = fmaxf(rm, __shfl_xor(rm, 8, 32));
      float mnew = fmaxf(mrow[i], rm);
      alpha[i] = expf(mrow[i] - mnew);
      mrow[i] = mnew;
      float p = expf(sv - mnew);
      pv[i] = p;
      float rs = p;
      rs += __shfl_xor(rs, 1, 32);
      rs += __shfl_xor(rs, 2, 32);
      rs += __shfl_xor(rs, 4, 32);
      rs += __shfl_xor(rs, 8, 32);
      lrow[i] = lrow[i] * alpha[i] + rs;
    }

    // Transpose P through the wave-private LDS patch. LDS ops from one wave
    // execute in order (ISA 7.3), so a dscnt wait + compiler fence is enough.
#pragma unroll
    for (int i = 0; i < 8; ++i) pt[wave][i + 8 * kh][lane15] = pv[i];
#pragma unroll
    for (int nt = 0; nt < 8; ++nt)
#pragma unroll
      for (int i = 0; i < 8; ++i) o[nt][i] *= alpha[i];
    wait_ds0();

    // O += P @ V  (16x128, K=16 -> 4 ksteps x 8 n-tiles)
#pragma unroll
    for (int ks = 0; ks < 4; ++ks) {
      v2f a;
      a.x = pt[wave][lane15][4 * ks + 2 * kh];
      a.y = pt[wave][lane15][4 * ks + 2 * kh + 1];
      const int kk = 4 * ks + 2 * kh;
#pragma unroll
      for (int nt = 0; nt < 8; ++nt) {
        v2f bv;  // B[k][n] = V[k][n]
        bv.x = vt[cur][kk][nt * 16 + lane15];
        bv.y = vt[cur][kk + 1][nt * 16 + lane15];
        o[nt] = __builtin_amdgcn_wmma_f32_16x16x4_f32(false, a, false, bv,
                                                      (short)0, o[nt],
                                                      false, false);
      }
    }

    wait_async0();     // my next-tile async copies have landed
    __syncthreads();   // everyone done reading buf[cur]; buf[cur^1] ready
    cur ^= 1;
  }

  // Normalize and write O (B,S,H,128)
#pragma unroll
  for (int nt = 0; nt < 8; ++nt)
#pragma unroll
    for (int i = 0; i < 8; ++i) {
      int r = qrow0 + i + 8 * kh;
      float val = o[nt][i] / lrow[i];
      ob[(((size_t)(b * S) + r) * H + h) * 128 + nt * 16 + lane15] = val;
    }
}

// ---------------------------------------------------------------------------
// attn = (o1 - lam*o2) * rsqrt(mean^2 + eps) * gamma * (1 - LAMBDA_INIT)
// One 128-thread block per (b,t,h).
// ---------------------------------------------------------------------------
__global__ __launch_bounds__(128) void combine_kernel(const float* __restrict__ o1,
                                                      const float* __restrict__ o2,
                                                      const float* __restrict__ lamp,
                                                      const float* __restrict__ gamma,
                                                      float* __restrict__ y) {
  __shared__ float red[4];
  const int e = threadIdx.x;
  const float lam = *lamp;
  const size_t off = (size_t)blockIdx.x * 128 + e;
  float a = o1[off] - lam * o2[off];
  float ss = a * a;
  ss += __shfl_xor(ss, 1, 32);
  ss += __shfl_xor(ss, 2, 32);
  ss += __shfl_xor(ss, 4, 32);
  ss += __shfl_xor(ss, 8, 32);
  ss += __shfl_xor(ss, 16, 32);
  if ((e & 31) == 0) red[e >> 5] = ss;
  __syncthreads();
  float tot = red[0] + red[1] + red[2] + red[3];
  float rms = rsqrtf(tot * (1.0f / 128.0f) + EPSV);
  y[off] = a * rms * gamma[e] * (1.0f - LAMBDA_INIT);
}

// ---------------------------------------------------------------------------
extern "C" void kernel_launch(void* const* d_in, const int* in_sizes, int n_in,
                              void* d_out, int out_size, void* d_ws, size_t ws_size,
                              hipStream_t stream) {
  const float* x    = (const float*)d_in[0];
  const float* Wq   = (const float*)d_in[1];
  const float* Wk   = (const float*)d_in[2];
  const float* Wv   = (const float*)d_in[3];
  const float* Wo   = (const float*)d_in[4];
  const float* lq1  = (const float*)d_in[5];
  const float* lk1  = (const float*)d_in[6];
  const float* lq2  = (const float*)d_in[7];
  const float* lk2  = (const float*)d_in[8];
  const float* gam  = (const float*)d_in[9];

  const int Bv = 2, Sv = 2048, Ev = 2048, Hv = 16;
  const int M = Bv * Sv;              // 4096
  const size_t nBSE = (size_t)M * Ev; // 8M floats

  float* qbuf = (float*)d_ws;
  float* kbuf = qbuf + nBSE;
  float* vbuf = kbuf + nBSE;
  float* o1   = vbuf + nBSE;
  float* o2   = o1 + nBSE;
  float* lamp = o2 + nBSE;   // 1 float
  float* ybuf = qbuf;        // q is dead after attention -> reuse

  dim3 gblk(Ev / 64, M / 128);
  proj_gemm<<<gblk, 256, 0, stream>>>(x, Wq, qbuf, M, Ev, Ev);
  proj_gemm<<<gblk, 256, 0, stream>>>(x, Wk, kbuf, M, Ev, Ev);
  proj_gemm<<<gblk, 256, 0, stream>>>(x, Wv, vbuf, M, Ev, Ev);

  const int ropeTotal = M * 32 * 32;  // (B*S) * 2H * HD/2
  rope_kernel<<<ropeTotal / 256, 256, 0, stream>>>(qbuf, Sv, ropeTotal);
  rope_kernel<<<ropeTotal / 256, 256, 0, stream>>>(kbuf, Sv, ropeTotal);

  lam_kernel<<<1, 1, 0, stream>>>(lq1, lk1, lq2, lk2, lamp);

  dim3 gattn(Sv / 128, Hv, Bv * 2);
  attn_kernel<<<gattn, 256, 0, stream>>>(qbuf, kbuf, vbuf, o1, o2);

  combine_kernel<<<M * Hv, 128, 0, stream>>>(o1, o2, lamp, gam, ybuf);

  proj_gemm<<<gblk, 256, 0, stream>>>(ybuf, Wo, (float*)d_out, M, Ev, Ev);
}